// TimeAttender_58695023067454
// MI455X (gfx1250) — compile-verified
//
#include <hip/hip_runtime.h>
#include <math.h>

typedef __attribute__((ext_vector_type(2))) float v2f;
typedef __attribute__((ext_vector_type(4))) float v4f;
typedef __attribute__((ext_vector_type(8))) float v8f;

#define S_LEN 4096
#define B_SZ  16
#define H2D   1024
#define NCH   32            // s-chunks for the weighted-sum pass
#define CHS   (S_LEN / NCH) // 128 timesteps per chunk

// ---------------------------------------------------------------------------
// Kernel 1: altered_state[b][n] = attn_b[n] + sum_k state[b][k] * attn_w[n][k]
// 16x1024x1024 f32 GEMM via V_WMMA_F32_16X16X4_F32 (M = batch = 16).
// One wave per 16-wide output column tile; 64 blocks total.
// ---------------------------------------------------------------------------
__global__ __launch_bounds__(32) void attend_gemm_wmma(
    const float* __restrict__ state, const float* __restrict__ attn_w,
    const float* __restrict__ attn_b, float* __restrict__ altered)
{
    const int lane = threadIdx.x;
    const int n0   = blockIdx.x * 16;
    const int half = lane >> 4;     // 0: K pair {0,1}; 1: K pair {2,3}
    const int l16  = lane & 15;

    v8f c = {};
    for (int k0 = 0; k0 < H2D; k0 += 4) {
        const int ka = k0 + half * 2;
        v2f a, b;
        a.x = state[l16 * H2D + ka];            // A[m=l16][ka]
        a.y = state[l16 * H2D + ka + 1];
        b.x = attn_w[(n0 + l16) * H2D + ka];    // B[ka][n=l16] = attn_w[n][ka]
        b.y = attn_w[(n0 + l16) * H2D + ka + 1];
        c = __builtin_amdgcn_wmma_f32_16x16x4_f32(
                false, a, false, b, (short)0, c, false, false);
    }

    const float bias = attn_b[n0 + l16];
#pragma unroll
    for (int i = 0; i < 8; ++i) {
        const int row = i + half * 8;
        altered[row * H2D + n0 + l16] = c[i] + bias;
    }
}

// ---------------------------------------------------------------------------
// Kernel 2: weights[b][s] = dot(altered[b][:], enc[s][b][:])  (1024-dim dot)
// One block per s; wave w handles batch b=w; b128 coalesced loads
// (512 B contiguous per wave per step) + wave32 shuffle reduction.
// ---------------------------------------------------------------------------
__global__ __launch_bounds__(512) void weights_kernel(
    const float* __restrict__ enc, const float* __restrict__ altered,
    float* __restrict__ weights)
{
    const int s    = blockIdx.x;
    const int b    = threadIdx.x >> 5;
    const int lane = threadIdx.x & 31;

    const float* __restrict__ er = enc + ((size_t)s * B_SZ + b) * H2D;
    const float* __restrict__ ar = altered + b * H2D;

    float acc = 0.0f;
#pragma unroll
    for (int i = 0; i < 8; ++i) {
        const int d = (lane + i * 32) * 4;          // covers 0..1023
        const v4f e = *(const v4f*)(er + d);
        const v4f a = *(const v4f*)(ar + d);
        acc += e.x * a.x + e.y * a.y + e.z * a.z + e.w * a.w;
    }
#pragma unroll
    for (int off = 16; off >= 1; off >>= 1)
        acc += __shfl_xor(acc, off, 32);

    if (lane == 0) weights[b * S_LEN + s] = acc;
}

// ---------------------------------------------------------------------------
// Kernel 3: normalized_weights = softmax(weights, axis=1). One block per b.
// ---------------------------------------------------------------------------
__global__ __launch_bounds__(256) void softmax_kernel(
    const float* __restrict__ weights, float* __restrict__ nw)
{
    __shared__ float red[256];
    const int b = blockIdx.x;
    const int t = threadIdx.x;
    const float* __restrict__ wr = weights + b * S_LEN;

    float v[16];
    float lmax = -INFINITY;
#pragma unroll
    for (int i = 0; i < 16; ++i) {
        v[i] = wr[t + i * 256];
        lmax = fmaxf(lmax, v[i]);
    }
    red[t] = lmax; __syncthreads();
    for (int off = 128; off >= 1; off >>= 1) {
        if (t < off) red[t] = fmaxf(red[t], red[t + off]);
        __syncthreads();
    }
    const float m = red[0]; __syncthreads();

    float lsum = 0.0f;
#pragma unroll
    for (int i = 0; i < 16; ++i) { v[i] = __expf(v[i] - m); lsum += v[i]; }
    red[t] = lsum; __syncthreads();
    for (int off = 128; off >= 1; off >>= 1) {
        if (t < off) red[t] += red[t + off];
        __syncthreads();
    }
    const float inv = 1.0f / red[0];
#pragma unroll
    for (int i = 0; i < 16; ++i) nw[b * S_LEN + t + i * 256] = v[i] * inv;
}

// ---------------------------------------------------------------------------
// Kernel 4: partial[c][b][d] = sum_{s in chunk c} nw[b][s] * enc[s][b][d]
// grid = (B_SZ, NCH) x 256 threads. Each thread owns 4 consecutive d values
// -> global_load_b128, 4 KB contiguous per block per timestep. The chunk's
// softmax weights are staged once into LDS. enc is streamed exactly once.
// ---------------------------------------------------------------------------
__global__ __launch_bounds__(256) void apply_partial_kernel(
    const float* __restrict__ enc, const float* __restrict__ nw,
    float* __restrict__ part)
{
    __shared__ float wsh[CHS];
    const int b  = blockIdx.x;
    const int c  = blockIdx.y;
    const int s0 = c * CHS;
    const int d  = threadIdx.x * 4;                 // 4 floats per thread, 0..1023

    if (threadIdx.x < CHS) wsh[threadIdx.x] = nw[b * S_LEN + s0 + threadIdx.x];
    __syncthreads();

    v4f acc = {};
    for (int i = 0; i < CHS; ++i) {
        const int s = s0 + i;
        const float* __restrict__ er = enc + ((size_t)s * B_SZ + b) * H2D + d;
        // hint the 64KB-strided stream a few iterations ahead
        __builtin_prefetch(er + (size_t)4 * B_SZ * H2D, 0, 0);
        const v4f e = *(const v4f*)er;
        const float w = wsh[i];
        acc.x += w * e.x; acc.y += w * e.y; acc.z += w * e.z; acc.w += w * e.w;
    }
    *(v4f*)(part + ((size_t)c * B_SZ + b) * H2D + d) = acc;
}

// ---------------------------------------------------------------------------
// Kernel 5: attention_applied[b][d] = sum_c partial[c][b][d] (deterministic)
// ---------------------------------------------------------------------------
__global__ __launch_bounds__(256) void reduce_kernel(
    const float* __restrict__ part, float* __restrict__ out)
{
    const int idx = blockIdx.x * 256 + threadIdx.x;   // b*1024 + d, 0..16383
    float acc = 0.0f;
#pragma unroll
    for (int c = 0; c < NCH; ++c) acc += part[c * (B_SZ * H2D) + idx];
    out[idx] = acc;
}

extern "C" void kernel_launch(void* const* d_in, const int* in_sizes, int n_in,
                              void* d_out, int out_size, void* d_ws, size_t ws_size,
                              hipStream_t stream) {
    (void)in_sizes; (void)n_in; (void)out_size; (void)ws_size;
    const float* enc    = (const float*)d_in[0];   // [S,B,2H]
    const float* state  = (const float*)d_in[1];   // [B,2H]
    // d_in[2] previous_attention, d_in[5] time_w, d_in[6] time_b: dead branch, skipped
    const float* attn_w = (const float*)d_in[3];   // [2H,2H]
    const float* attn_b = (const float*)d_in[4];   // [2H]

    float* out = (float*)d_out;
    float* attention_applied = out;                 // [B,2H] = 16384 floats
    float* nw                = out + B_SZ * H2D;    // [B,S]  = 65536 floats

    float* ws      = (float*)d_ws;
    float* altered = ws;                                    // 16*1024
    float* weights = ws + B_SZ * H2D;                       // 16*4096
    float* part    = ws + B_SZ * H2D + B_SZ * S_LEN;        // NCH*16*1024

    attend_gemm_wmma    <<<64,   32, 0, stream>>>(state, attn_w, attn_b, altered);
    weights_kernel      <<<S_LEN, 512, 0, stream>>>(enc, altered, weights);
    softmax_kernel      <<<B_SZ, 256, 0, stream>>>(weights, nw);
    apply_partial_kernel<<<dim3(B_SZ, NCH), 256, 0, stream>>>(enc, nw, part);
    reduce_kernel       <<<64,  256, 0, stream>>>(part, attention_applied);
}